// LCNSpikingHybrid_4698694222620
// MI455X (gfx1250) — compile-verified
//
#include <hip/hip_runtime.h>
#include <hip/hip_bf16.h>

// ---------------- problem constants (from reference) ----------------
#define ALPHA 0.9f
#define BETA  0.8f
#define KNN_K 25
#define BATCH 128
#define TSTEPS 10
#define D0 14400
#define D1 7200
#define D2 3600
#define D3 1800
#define D4 900
#define D5 450
#define NOUT 2

#define DPB 8            // output neurons per block in gather kernels

typedef float v2f __attribute__((ext_vector_type(2)));
typedef float v8f __attribute__((ext_vector_type(8)));

// ---------------- zero-init ----------------
__global__ void zero_kernel(float* __restrict__ p, int n) {
    int i = blockIdx.x * blockDim.x + threadIdx.x;
    int stride = gridDim.x * blockDim.x;
    for (; i < n; i += stride) p[i] = 0.0f;
}

// ---------------- transpose one timestep: (B,T,D0) slice t -> (D0,B) ----------------
__global__ void transpose_t_kernel(const float* __restrict__ in, float* __restrict__ out, int t) {
    __shared__ float tile[32][33];
    int d0 = blockIdx.x * 32;          // D0/32 = 450 blocks
    int b0 = blockIdx.y * 32;          // B/32  = 4 blocks
    int tx = threadIdx.x, ty = threadIdx.y;   // block (32, 8)
    #pragma unroll
    for (int j = 0; j < 32; j += 8) {
        int b = b0 + ty + j;
        tile[ty + j][tx] = in[((size_t)b * TSTEPS + t) * D0 + d0 + tx];  // coalesced over d
    }
    __syncthreads();
    #pragma unroll
    for (int j = 0; j < 32; j += 8) {
        int d = d0 + ty + j;
        out[(size_t)d * BATCH + b0 + tx] = tile[tx][ty + j];             // coalesced over b
    }
}

// ---------------- LIF gather layer: cur = gather-dot; update syn/mem; optional spike out ----
// xin: (Dprev, BATCH); syn/mem/spk: (D, BATCH); w/knn: (D, K); bias/thr: (D)
__global__ void lcn_lif_kernel(const float* __restrict__ xin, const float* __restrict__ w,
                               const float* __restrict__ bias, const int* __restrict__ knn,
                               const float* __restrict__ thr,
                               float* __restrict__ syn, float* __restrict__ mem,
                               float* __restrict__ spk, int D) {
    __shared__ int   s_idx[DPB * KNN_K];
    __shared__ float s_w  [DPB * KNN_K];
    const int b    = threadIdx.x;             // lane = batch element (4 waves of 32)
    const int dbas = blockIdx.x * DPB;
    const int base = dbas * KNN_K;
    for (int i = threadIdx.x; i < DPB * KNN_K; i += BATCH) {
        if (base + i < D * KNN_K) { s_idx[i] = knn[base + i]; s_w[i] = w[base + i]; }
    }
    __syncthreads();
    for (int j = 0; j < DPB; ++j) {
        int d = dbas + j;
        if (d >= D) break;
        if (j + 1 < DPB)   // CDNA5 global_prefetch_b8 of next neuron's first gather row
            __builtin_prefetch(&xin[(size_t)s_idx[(j + 1) * KNN_K] * BATCH + b], 0, 0);
        const int*   ip = &s_idx[j * KNN_K];
        const float* wp = &s_w[j * KNN_K];
        float acc = bias[d];
        #pragma unroll
        for (int k = 0; k < KNN_K; ++k)
            acc = fmaf(xin[(size_t)ip[k] * BATCH + b], wp[k], acc);   // 512B coalesced bursts
        const size_t o = (size_t)d * BATCH + b;
        const float th = thr[d];
        const float m0 = mem[o];
        const float reset = (m0 > th) ? 1.0f : 0.0f;   // heaviside(mem_prev - thr)
        const float s = ALPHA * syn[o] + acc;
        const float m = BETA * m0 + s - reset * th;
        syn[o] = s;
        mem[o] = m;
        if (spk) spk[o] = (m > th) ? 1.0f : 0.0f;      // heaviside(mem_new - thr)
    }
}

// ---------------- feed-forward gather layer + ReLU ----------------
__global__ void lcn_relu_kernel(const float* __restrict__ xin, const float* __restrict__ w,
                                const float* __restrict__ bias, const int* __restrict__ knn,
                                float* __restrict__ out, int D) {
    __shared__ int   s_idx[DPB * KNN_K];
    __shared__ float s_w  [DPB * KNN_K];
    const int b    = threadIdx.x;
    const int dbas = blockIdx.x * DPB;
    const int base = dbas * KNN_K;
    for (int i = threadIdx.x; i < DPB * KNN_K; i += BATCH) {
        if (base + i < D * KNN_K) { s_idx[i] = knn[base + i]; s_w[i] = w[base + i]; }
    }
    __syncthreads();
    for (int j = 0; j < DPB; ++j) {
        int d = dbas + j;
        if (d >= D) break;
        if (j + 1 < DPB && dbas + j + 1 < D)
            __builtin_prefetch(&xin[(size_t)s_idx[(j + 1) * KNN_K] * BATCH + b], 0, 0);
        const int*   ip = &s_idx[j * KNN_K];
        const float* wp = &s_w[j * KNN_K];
        float acc = bias[d];
        #pragma unroll
        for (int k = 0; k < KNN_K; ++k)
            acc = fmaf(xin[(size_t)ip[k] * BATCH + b], wp[k], acc);
        out[(size_t)d * BATCH + b] = fmaxf(acc, 0.0f);
    }
}

// ---------------- final FC via V_WMMA_F32_16X16X4_F32 ----------------
// x: (D5, BATCH) col-batch layout; fcW: (NOUT, D5) row-major; out: (BATCH, NOUT) row-major.
// One block of 256 threads = 8 wave32; wave w computes batch rows [16w, 16w+16).
// A lane layout (16x4 f32): lane L -> M = L&15, K-pair = 2*(L>>4) .. +1 (VGPR0/1).
// B lane layout (4x16 f32): lane L -> N = L&15, same K-pair striping.
// C/D (16x16 f32): lane L -> N = L&15, VGPR r -> M = r + 8*(L>>4).
// Padding is arithmetic (clamped addresses * 0/1 masks), never EXEC branches,
// so the steady-state loop is 4 unconditional global_load_b32 + v_wmma.
__global__ void fc_wmma_kernel(const float* __restrict__ x, const float* __restrict__ fcW,
                               const float* __restrict__ fcb, float* __restrict__ out) {
    const int wave  = threadIdx.x >> 5;   // wave32
    const int lane  = threadIdx.x & 31;
    const int half  = lane >> 4;          // 0: K {0,1}; 1: K {2,3}
    const int lm    = lane & 15;          // M for A, N for B/D
    const int mbase = wave * 16;
    const float bmask = (lm < NOUT) ? 1.0f : 0.0f;    // zero-pad N beyond 2
    const int   brow  = (lm < NOUT) ? lm : (NOUT - 1);
    const float* __restrict__ xa = x + (size_t)(half * 2) * BATCH + mbase + lm;
    const float* __restrict__ wb = fcW + (size_t)brow * D5 + half * 2;

    v8f c = {};
    // 112 full K=4 steps cover k = 0..447 with every address valid: no conditionals.
    for (int k0 = 0; k0 < 448; k0 += 4) {
        v2f a, bb;
        a.x  = xa[(size_t)(k0 + 0) * BATCH];
        a.y  = xa[(size_t)(k0 + 1) * BATCH];
        bb.x = wb[k0 + 0] * bmask;
        bb.y = wb[k0 + 1] * bmask;
        c = __builtin_amdgcn_wmma_f32_16x16x4_f32(false, a, false, bb, (short)0, c, false, false);
    }
    // Tail: k = 448,449 live in half==0 lanes; half==1 lanes (k=450,451) are masked zeros
    // loaded from a clamped valid address (select + multiply, no EXEC divergence).
    {
        const float kmask = (half == 0) ? 1.0f : 0.0f;
        const int   kt    = (half == 0) ? 448 : 446;    // clamped, always in range
        v2f a, bb;
        a.x  = x[(size_t)(kt + 0) * BATCH + mbase + lm] * kmask;
        a.y  = x[(size_t)(kt + 1) * BATCH + mbase + lm] * kmask;
        bb.x = fcW[(size_t)brow * D5 + kt + 0] * (bmask * kmask);
        bb.y = fcW[(size_t)brow * D5 + kt + 1] * (bmask * kmask);
        c = __builtin_amdgcn_wmma_f32_16x16x4_f32(false, a, false, bb, (short)0, c, false, false);
    }
    if (lm < NOUT) {
        const float bias = fcb[lm];
        #pragma unroll
        for (int r = 0; r < 8; ++r) {
            const int m = mbase + r + half * 8;
            out[m * NOUT + lm] = c[r] + bias;
        }
    }
}

// ---------------- host-side orchestration ----------------
extern "C" void kernel_launch(void* const* d_in, const int* in_sizes, int n_in,
                              void* d_out, int out_size, void* d_ws, size_t ws_size,
                              hipStream_t stream) {
    (void)in_sizes; (void)n_in; (void)out_size; (void)ws_size;
    // setup_inputs() dict order:
    const float* input = (const float*)d_in[0];
    const float* w0    = (const float*)d_in[1];
    const float* b0    = (const float*)d_in[2];
    const int*   knn0  = (const int*)  d_in[3];
    const float* w1    = (const float*)d_in[4];
    const float* b1    = (const float*)d_in[5];
    const int*   knn1  = (const int*)  d_in[6];
    const float* w2    = (const float*)d_in[7];
    const float* b2    = (const float*)d_in[8];
    const int*   knn2  = (const int*)  d_in[9];
    const float* w3    = (const float*)d_in[10];
    const float* b3    = (const float*)d_in[11];
    const int*   knn3  = (const int*)  d_in[12];
    const float* w4    = (const float*)d_in[13];
    const float* b4    = (const float*)d_in[14];
    const int*   knn4  = (const int*)  d_in[15];
    const float* thr0  = (const float*)d_in[16];
    const float* thr1  = (const float*)d_in[17];
    const float* fcW   = (const float*)d_in[18];
    const float* fcb   = (const float*)d_in[19];
    float* out = (float*)d_out;

    // workspace layout (floats), all (D, BATCH) column-batch tensors
    float* ws   = (float*)d_ws;
    float* xT   = ws;                                  // D0*BATCH = 1,843,200
    float* syn0 = xT   + (size_t)D0 * BATCH;           // D1*BATCH
    float* mem0 = syn0 + (size_t)D1 * BATCH;
    float* syn1 = mem0 + (size_t)D1 * BATCH;           // D2*BATCH
    float* mem1 = syn1 + (size_t)D2 * BATCH;
    float* spk0 = mem1 + (size_t)D2 * BATCH;           // D1*BATCH
    float* a2   = spk0 + (size_t)D1 * BATCH;           // D3*BATCH
    float* a3   = a2   + (size_t)D3 * BATCH;           // D4*BATCH
    float* a4   = a3   + (size_t)D4 * BATCH;           // D5*BATCH

    // zero LIF state (syn0,mem0,syn1,mem1 are contiguous)
    const int nstate = 2 * (D1 * BATCH) + 2 * (D2 * BATCH);
    zero_kernel<<<1024, 256, 0, stream>>>(syn0, nstate);

    dim3 tb(32, 8), tg(D0 / 32, BATCH / 32);
    const int g0 = (D1 + DPB - 1) / DPB;   // 900
    const int g1 = (D2 + DPB - 1) / DPB;   // 450
    for (int t = 0; t < TSTEPS; ++t) {
        transpose_t_kernel<<<tg, tb, 0, stream>>>(input, xT, t);
        lcn_lif_kernel<<<g0, BATCH, 0, stream>>>(xT,   w0, b0, knn0, thr0, syn0, mem0, spk0,    D1);
        lcn_lif_kernel<<<g1, BATCH, 0, stream>>>(spk0, w1, b1, knn1, thr1, syn1, mem1, nullptr, D2);
    }

    lcn_relu_kernel<<<(D3 + DPB - 1) / DPB, BATCH, 0, stream>>>(mem1, w2, b2, knn2, a2, D3);
    lcn_relu_kernel<<<(D4 + DPB - 1) / DPB, BATCH, 0, stream>>>(a2,   w3, b3, knn3, a3, D4);
    lcn_relu_kernel<<<(D5 + DPB - 1) / DPB, BATCH, 0, stream>>>(a3,   w4, b4, knn4, a4, D5);

    fc_wmma_kernel<<<1, 256, 0, stream>>>(a4, fcW, fcb, out);
}